// TripleSig_56169582297405
// MI455X (gfx1250) — compile-verified
//
#include <hip/hip_runtime.h>

// Problem constants (match reference setup)
#define N_J   25
#define N_A   3
#define N_P   2300
#define BLK   256

typedef __attribute__((ext_vector_type(4))) unsigned int v4u;
typedef __attribute__((ext_vector_type(8))) int          v8i;
typedef __attribute__((ext_vector_type(4))) int          v4i;
typedef __attribute__((ext_vector_type(4))) float        f4;

// Issue one TDM 1-D tile load (nelem 32-bit elements, nelem <= 65535) from
// global `gaddr` into LDS byte offset `lds_off`. Descriptor layout per CDNA5
// ISA ch.8 (Tensor DMA Descriptor). Wave-level op, tracked by TENSORcnt.
__device__ __forceinline__ void tdm_load_1d_b32(unsigned lds_off,
                                                unsigned long long gaddr,
                                                unsigned nelem) {
  // --- Group 0 (128b): count=1 | lds_addr | global_addr | type=2 ---
  v4u g0;
  g0.x = 1u;                                   // count=1 (valid), user mode
  g0.y = lds_off;                              // LDS byte address
  g0.z = (unsigned)(gaddr & 0xFFFFFFFFull);    // global_addr[31:0]
  g0.w = (unsigned)(gaddr >> 32) | 0x80000000u;// global_addr[56:32] | type=2<<30

  // --- Group 1 (256b) ---
  v8i g1;
  g1[0] = 0x00020000;                 // workgroup_mask=0, data_size=2 (4B)
  g1[1] = (int)((nelem & 0xFFFFu) << 16);        // tensor_dim0[15:0] @ bits 63:48
  g1[2] = (int)((nelem >> 16) | (1u << 16));     // tensor_dim0[31:16] | tensor_dim1=1
  g1[3] = (int)((nelem & 0xFFFFu) << 16);        // tensor_dim1 hi=0 | tile_dim0=nelem
  g1[4] = 1;                                     // tile_dim1=1, tile_dim2=0
  g1[5] = (int)nelem;                            // tensor_dim0_stride[31:0]
  g1[6] = 0;                                     // stride0 hi, stride1 lo
  g1[7] = 0;                                     // stride1 hi

  v4i z4 = {0, 0, 0, 0};                         // groups 2/3 unused (<=2D)
  v8i z8 = {0, 0, 0, 0, 0, 0, 0, 0};
  __builtin_amdgcn_tensor_load_to_lds(g0, g1, z4, z4, z8, 0);
}

__global__ void __launch_bounds__(BLK)
triple_sig_kernel(const float* __restrict__ x,   // [B,T,3,25]
                  const int*   __restrict__ tri, // [2300,3]
                  float*       __restrict__ out) // [B,T,2300,12]
{
  __shared__ float sx[N_A * N_J];     // 300 B: x slice for this (b,t)
  __shared__ int   stri[N_P * 3];     // 27.6 KB: triples table

  const int bt = blockIdx.x;          // flattened (b,t)

  // ---- Stage inputs into LDS via the Tensor Data Mover (wave 0 only) ----
  if ((threadIdx.x >> 5) == 0) {
    unsigned sx_off  = (unsigned)(unsigned long long)(uintptr_t)&sx[0];
    unsigned tri_off = (unsigned)(unsigned long long)(uintptr_t)&stri[0];
    unsigned long long xg = (unsigned long long)(uintptr_t)(x + (size_t)bt * (N_A * N_J));
    unsigned long long tg = (unsigned long long)(uintptr_t)tri;
    tdm_load_1d_b32(sx_off,  xg, N_A * N_J);   // 75 f32
    tdm_load_1d_b32(tri_off, tg, N_P * 3);     // 6900 i32
    __builtin_amdgcn_s_wait_tensorcnt(0);
  }
  __syncthreads();

  // ---- Per-triple degree-2 path signature; 3x b128 non-temporal stores ----
  for (int p = threadIdx.x; p < N_P; p += BLK) {
    const int j0 = stri[3 * p + 0];
    const int j1 = stri[3 * p + 1];
    const int j2 = stri[3 * p + 2];

    float d1[3], d2[3];
#pragma unroll
    for (int a = 0; a < 3; ++a) {
      const float p0 = sx[a * N_J + j0];
      const float p1 = sx[a * N_J + j1];
      const float p2 = sx[a * N_J + j2];
      d1[a] = p1 - p0;
      d2[a] = p2 - p1;
    }

    float o[12];
    o[0] = d1[0] + d2[0];
    o[1] = d1[1] + d2[1];
    o[2] = d1[2] + d2[2];
#pragma unroll
    for (int i = 0; i < 3; ++i)
#pragma unroll
      for (int j = 0; j < 3; ++j)
        o[3 + 3 * i + j] = 0.5f * (d1[i] * d1[j] + d2[i] * d2[j]) + d1[i] * d2[j];

    float* op = out + ((size_t)bt * N_P + p) * 12;   // 48B-aligned
    f4 v0 = {o[0], o[1], o[2],  o[3]};
    f4 v1 = {o[4], o[5], o[6],  o[7]};
    f4 v2 = {o[8], o[9], o[10], o[11]};
    __builtin_nontemporal_store(v0, (f4*)(op + 0));
    __builtin_nontemporal_store(v1, (f4*)(op + 4));
    __builtin_nontemporal_store(v2, (f4*)(op + 8));
  }
}

extern "C" void kernel_launch(void* const* d_in, const int* in_sizes, int n_in,
                              void* d_out, int out_size, void* d_ws, size_t ws_size,
                              hipStream_t stream) {
  const float* x   = (const float*)d_in[0];
  const int*   tri = (const int*)d_in[1];
  float*       out = (float*)d_out;

  const int BT = in_sizes[0] / (N_A * N_J);  // B*T = 2400
  triple_sig_kernel<<<dim3(BT), dim3(BLK), 0, stream>>>(x, tri, out);
}